// NasAzcsCell_44916767981746
// MI455X (gfx1250) — compile-verified
//
#include <hip/hip_runtime.h>

// ---------------------------------------------------------------------------
// Problem constants (from the reference)
// ---------------------------------------------------------------------------
#define N_NODES 50000
#define N_EDGES 800000
#define CUR     128
#define HID     96
#define OUT     128
#define NEG_SLOPE 0.01f

typedef __attribute__((ext_vector_type(2))) float v2f;
typedef __attribute__((ext_vector_type(8))) float v8f;

typedef unsigned int u32x4 __attribute__((ext_vector_type(4)));
typedef int          i32x8 __attribute__((ext_vector_type(8)));
typedef int          i32x4 __attribute__((ext_vector_type(4)));

// D = A(16x4 f32) * B(4x16 f32) + C(16x16 f32), full fp32 WMMA on CDNA5.
__device__ __forceinline__ v8f wmma4(v2f a, v2f b, v8f c) {
  return __builtin_amdgcn_wmma_f32_16x16x4_f32(
      /*neg_a=*/false, a, /*neg_b=*/false, b,
      /*c_mod=*/(short)0, c, /*reuse_a=*/false, /*reuse_b=*/false);
}

// ---------------------------------------------------------------------------
// TDM: DMA a 2D row-major f32 matrix [krows x ncol] from global into LDS.
// D# built per CDNA5 ISA ch.8: group0 = {flags, lds_addr, global_addr, type=2},
// group1 = {data_size=4B, tensor_dim0/1, tile_dim0/1, dim0_stride}.
// Issued by one wave; completion tracked with TENSORcnt.
// This toolchain exposes the 6-arg builtin:
//   (u32x4 g0, i32x8 g1, i32x4 g2, i32x4 g3, i32x8 extra, i32 cpol)
// ---------------------------------------------------------------------------
__device__ __forceinline__ void tdm_load_2d(float* lds_dst, const float* gsrc,
                                            unsigned ncol, unsigned krows) {
  unsigned long long ga = (unsigned long long)(size_t)gsrc;
  unsigned lds = (unsigned)(size_t)lds_dst;      // low 32 bits = LDS byte offset

  u32x4 g0;
  g0[0] = 1u;                                    // count=1 (valid user D#)
  g0[1] = lds;                                   // lds_addr [63:32]
  g0[2] = (unsigned)(ga & 0xFFFFFFFFu);          // global_addr [95:64]
  g0[3] = (unsigned)((ga >> 32) & 0x01FFFFFFu)   // global_addr [120:96]
        | (2u << 30);                            // type=2 ("image") [127:126]

  i32x8 g1;
  g1[0] = (int)(2u << 16);                       // data_size=2 (4 bytes), mask=0
  g1[1] = (int)(ncol  << 16);                    // tensor_dim0[15:0]  @bit48
  g1[2] = (int)(krows << 16);                    // tensor_dim1[15:0]  @bit80
  g1[3] = (int)(ncol  << 16);                    // tile_dim0          @bit112
  g1[4] = (int)krows;                            // tile_dim1          @bit128
  g1[5] = (int)ncol;                             // tensor_dim0_stride @bit160
  g1[6] = 0;
  g1[7] = 0;

  i32x4 z4 = {0, 0, 0, 0};                       // groups 2/3 unused (2D tensor)
  i32x8 z8 = {0, 0, 0, 0, 0, 0, 0, 0};
  __builtin_amdgcn_tensor_load_to_lds(g0, g1, z4, z4, z8, 0);
}

// ---------------------------------------------------------------------------
// Utility kernels
// ---------------------------------------------------------------------------
__global__ void k_zero(float* __restrict__ p, size_t n) {
  size_t i = (size_t)blockIdx.x * blockDim.x + threadIdx.x;
  size_t stride = (size_t)gridDim.x * blockDim.x;
  for (; i < n; i += stride) p[i] = 0.0f;
}

__global__ void k_copy4(const float4* __restrict__ src, float4* __restrict__ dst, size_t n4) {
  size_t i = (size_t)blockIdx.x * blockDim.x + threadIdx.x;
  size_t stride = (size_t)gridDim.x * blockDim.x;
  for (; i < n4; i += stride) dst[i] = src[i];
}

// ---------------------------------------------------------------------------
// WMMA GEMM: Y[N_NODES x NCOL] = A[N_NODES x K] * W[K x NCOL] (+ bias)
// - W staged once per workgroup into LDS by the Tensor Data Mover
// - one wave32 owns a 16-row strip x TNB 16-col tiles (A regs reused TNB times)
// A layout (f32 16x4): lanes 0-15 hold K=0,1 ; lanes 16-31 hold K=2,3
// C/D layout: VGPR r -> row r (lanes 0-15) / row r+8 (lanes 16-31), N = lane&15
// ---------------------------------------------------------------------------
template <int K, int NCOL, int TNB, bool HAS_BIAS>
__global__ void k_gemm(const float* __restrict__ A, const float* __restrict__ W,
                       const float* __restrict__ bias, float* __restrict__ Y) {
  extern __shared__ float sW[];                  // K * NCOL floats
  if ((threadIdx.x >> 5) == 0) {                 // wave 0 issues the DMA
    tdm_load_2d(sW, W, NCOL, K);
    __builtin_amdgcn_s_wait_tensorcnt(0);
  }
  __syncthreads();

  const int TGROUPS = NCOL / 16 / TNB;
  const int total = (N_NODES / 16) * TGROUPS;
  int wave = blockIdx.x * (blockDim.x >> 5) + (threadIdx.x >> 5);
  if (wave >= total) return;
  int strip = wave / TGROUPS;                    // 16-row strip of A
  int tg    = wave % TGROUPS;                    // group of TNB column tiles
  int lane = threadIdx.x & 31;
  int half = lane >> 4;
  int l15  = lane & 15;
  int row  = strip * 16 + l15;
  int col0 = tg * TNB * 16 + l15;

  v8f acc[TNB];
#pragma unroll
  for (int t = 0; t < TNB; ++t) acc[t] = (v8f){};

  const float* ar = A + (size_t)row * K;
#pragma unroll 2
  for (int k = 0; k < K; k += 4) {
    int ka = k + half * 2;
    v2f a; a.x = ar[ka]; a.y = ar[ka + 1];
#pragma unroll
    for (int t = 0; t < TNB; ++t) {
      int col = col0 + t * 16;
      v2f b; b.x = sW[ka * NCOL + col]; b.y = sW[(ka + 1) * NCOL + col];
      acc[t] = wmma4(a, b, acc[t]);
    }
  }

#pragma unroll
  for (int t = 0; t < TNB; ++t) {
    int col = col0 + t * 16;
    float bv = HAS_BIAS ? bias[col] : 0.0f;
#pragma unroll
    for (int r = 0; r < 8; ++r) {
      int m = strip * 16 + r + 8 * half;
      Y[(size_t)m * NCOL + col] = acc[t][r] + bv;
    }
  }
}

// ---------------------------------------------------------------------------
// Edge kernels (scatter via global_atomic_add_f32; accumulators are L2-resident)
// ---------------------------------------------------------------------------
__global__ void k_deg(const long long* __restrict__ ei, const float* __restrict__ ew,
                      float* __restrict__ deg, float* __restrict__ cnt) {
  int e = blockIdx.x * blockDim.x + threadIdx.x;
  if (e >= N_EDGES) return;
  int d = (int)ei[N_EDGES + e];
  float w = ew[e];
  atomicAdd(&deg[d], w);
  atomicAdd(&cnt[d], 1.0f);
}

__global__ void k_dinv(const float* __restrict__ deg, float* __restrict__ dinv) {
  int i = blockIdx.x * blockDim.x + threadIdx.x;
  if (i >= N_NODES) return;
  float dg = deg[i];
  dinv[i] = (dg > 0.0f) ? rsqrtf(fmaxf(dg, 1e-30f)) : 0.0f;
}

// one wave32 per edge; 96 features = 3 per lane
__global__ void k_aggr(const long long* __restrict__ ei, const float* __restrict__ ew,
                       const float* __restrict__ xh, float* __restrict__ aggr) {
  int e = blockIdx.x * (blockDim.x >> 5) + (threadIdx.x >> 5);
  int lane = threadIdx.x & 31;
  if (e >= N_EDGES) return;
  int s = (int)ei[e];
  int d = (int)ei[N_EDGES + e];
  float w = ew[e];
  const float* xs = xh + (size_t)s * HID;
  float* ag = aggr + (size_t)d * HID;
#pragma unroll
  for (int f = lane; f < HID; f += 32) atomicAdd(&ag[f], xs[f] * w);
}

// one wave32 per edge; 128 features = 4 per lane; symmetric-norm weighting
__global__ void k_prop(const long long* __restrict__ ei, const float* __restrict__ ew,
                       const float* __restrict__ dinv, const float* __restrict__ xw,
                       float* __restrict__ prop) {
  int e = blockIdx.x * (blockDim.x >> 5) + (threadIdx.x >> 5);
  int lane = threadIdx.x & 31;
  if (e >= N_EDGES) return;
  int s = (int)ei[e];
  int d = (int)ei[N_EDGES + e];
  float nrm = dinv[s] * ew[e] * dinv[d];
  const float* xs = xw + (size_t)s * OUT;
  float* pr = prop + (size_t)d * OUT;
#pragma unroll
  for (int f = lane; f < OUT; f += 32) atomicAdd(&pr[f], xs[f] * nrm);
}

// ---------------------------------------------------------------------------
// Fused finale. Per wave: one 16-row strip x 4 column tiles of o3.
//   o_sage = mean@Wl + bl + xh@Wr     (C seeded with bl)
//   o_arma = relu(prop + xh@V + ab)   (C seeded with prop + ab)
//   o3     = relu(leaky(o_sage) + leaky(o_arma))
// All three weight matrices staged via TDM into 144 KB of LDS.
// ---------------------------------------------------------------------------
__global__ void k_final(const float* __restrict__ xh, const float* __restrict__ aggr,
                        const float* __restrict__ cnt, const float* __restrict__ prop,
                        const float* __restrict__ Wl, const float* __restrict__ bl,
                        const float* __restrict__ Wr, const float* __restrict__ Wv,
                        const float* __restrict__ ab, float* __restrict__ out) {
  extern __shared__ float smem[];                // 3 * HID*OUT floats
  float* sWl = smem;
  float* sWr = sWl + HID * OUT;
  float* sWv = sWr + HID * OUT;
  if ((threadIdx.x >> 5) == 0) {
    tdm_load_2d(sWl, Wl, OUT, HID);
    tdm_load_2d(sWr, Wr, OUT, HID);
    tdm_load_2d(sWv, Wv, OUT, HID);
    __builtin_amdgcn_s_wait_tensorcnt(0);
  }
  __syncthreads();

  const int TNB = 4;                             // column tiles per wave
  const int TGROUPS = OUT / 16 / TNB;            // = 2
  const int total = (N_NODES / 16) * TGROUPS;
  int wave = blockIdx.x * (blockDim.x >> 5) + (threadIdx.x >> 5);
  if (wave >= total) return;
  int strip = wave / TGROUPS;
  int tg    = wave % TGROUPS;
  int lane = threadIdx.x & 31;
  int half = lane >> 4;
  int l15  = lane & 15;
  int row  = strip * 16 + l15;
  int col0 = tg * TNB * 16 + l15;

  v8f acc_s[TNB], acc_a[TNB];
#pragma unroll
  for (int t = 0; t < TNB; ++t) {
    int col = col0 + t * 16;
    float bs = bl[col];
    float ba = ab[col];
#pragma unroll
    for (int r = 0; r < 8; ++r) {
      int m = strip * 16 + r + 8 * half;
      acc_s[t][r] = bs;                                    // sage bias
      acc_a[t][r] = prop[(size_t)m * OUT + col] + ba;      // arma: prop + bias
    }
  }

  float rc = 1.0f / fmaxf(cnt[row], 1.0f);       // mean normalizer for this row
  const float* xr = xh   + (size_t)row * HID;
  const float* ar = aggr + (size_t)row * HID;

  for (int k = 0; k < HID; k += 4) {
    int ka = k + half * 2;
    v2f am; am.x = ar[ka] * rc; am.y = ar[ka + 1] * rc;    // mean row
    v2f ax; ax.x = xr[ka];      ax.y = xr[ka + 1];         // xh row
#pragma unroll
    for (int t = 0; t < TNB; ++t) {
      int col = col0 + t * 16;
      v2f bwl; bwl.x = sWl[ka * OUT + col]; bwl.y = sWl[(ka + 1) * OUT + col];
      v2f bwr; bwr.x = sWr[ka * OUT + col]; bwr.y = sWr[(ka + 1) * OUT + col];
      v2f bwv; bwv.x = sWv[ka * OUT + col]; bwv.y = sWv[(ka + 1) * OUT + col];
      acc_s[t] = wmma4(am, bwl, acc_s[t]);                 // mean @ sage_Wl
      acc_s[t] = wmma4(ax, bwr, acc_s[t]);                 // + xh @ sage_Wr
      acc_a[t] = wmma4(ax, bwv, acc_a[t]);                 // prop + xh @ arma_V
    }
  }

#pragma unroll
  for (int t = 0; t < TNB; ++t) {
    int col = col0 + t * 16;
#pragma unroll
    for (int r = 0; r < 8; ++r) {
      int m = strip * 16 + r + 8 * half;
      float os = acc_s[t][r];
      float o1 = (os > 0.0f) ? os : NEG_SLOPE * os;        // leaky(o_sage)
      float oa = fmaxf(acc_a[t][r], 0.0f);                 // relu(o_arma)
      float o2 = (oa > 0.0f) ? oa : NEG_SLOPE * oa;        // leaky (id on >=0)
      float v = o1 + o2;
      out[(size_t)m * OUT + col] = fmaxf(v, 0.0f);         // final relu
    }
  }
}

// ---------------------------------------------------------------------------
// Launch
// ---------------------------------------------------------------------------
extern "C" void kernel_launch(void* const* d_in, const int* in_sizes, int n_in,
                              void* d_out, int out_size, void* d_ws, size_t ws_size,
                              hipStream_t stream) {
  (void)in_sizes; (void)n_in; (void)out_size; (void)ws_size;

  // inputs in setup_inputs() order (d_in[0] = h is unused: his = x)
  const float*     x   = (const float*)d_in[1];
  const long long* ei  = (const long long*)d_in[2];   // [2, E] int64: row0=src, row1=dst
  const float*     ew  = (const float*)d_in[3];
  const float*     pW  = (const float*)d_in[4];
  const float*     pb  = (const float*)d_in[5];
  const float*     sWl = (const float*)d_in[6];
  const float*     sbl = (const float*)d_in[7];
  const float*     sWr = (const float*)d_in[8];
  const float*     aW  = (const float*)d_in[9];
  const float*     aV  = (const float*)d_in[10];
  const float*     abv = (const float*)d_in[11];

  float* outF = (float*)d_out;             // [N*CUR] his  ++  [N*OUT] o3
  float* his  = outF;
  float* o3   = outF + (size_t)N_NODES * CUR;

  // workspace layout (floats)
  float* ws   = (float*)d_ws;
  float* xh   = ws;                                    // N*HID
  float* xw   = xh   + (size_t)N_NODES * HID;          // N*OUT
  float* aggr = xw   + (size_t)N_NODES * OUT;          // N*HID  (zeroed)
  float* prop = aggr + (size_t)N_NODES * HID;          // N*OUT  (zeroed)
  float* deg  = prop + (size_t)N_NODES * OUT;          // N      (zeroed)
  float* cnt  = deg  + (size_t)N_NODES;                // N      (zeroed)
  float* dinv = cnt  + (size_t)N_NODES;                // N      (fully written)

  const int B = 256;               // 8 wave32s per block
  const int WPB = B / 32;

  // 1) zero the atomic accumulators (aggr|prop|deg|cnt are contiguous)
  size_t nzero = (size_t)N_NODES * (HID + OUT + 2);
  k_zero<<<2048, B, 0, stream>>>(aggr, nzero);

  // 2) his = x
  size_t n4 = (size_t)N_NODES * CUR / 4;
  k_copy4<<<2048, B, 0, stream>>>((const float4*)x, (float4*)his, n4);

  // 3) xh = x @ pre_W + pre_b   (WMMA; wave owns all 6 col tiles; W via TDM)
  {
    const int total = (N_NODES / 16) * 1;              // TGROUPS = 96/16/6 = 1
    size_t shmem = (size_t)CUR * HID * sizeof(float);  // 48 KB
    k_gemm<CUR, HID, 6, true>
        <<<(total + WPB - 1) / WPB, B, shmem, stream>>>(x, pW, pb, xh);
  }

  // 4) degree + count
  k_deg<<<(N_EDGES + B - 1) / B, B, 0, stream>>>(ei, ew, deg, cnt);

  // 5) dinv = rsqrt(deg)
  k_dinv<<<(N_NODES + B - 1) / B, B, 0, stream>>>(deg, dinv);

  // 6) aggr[dst] += xh[src] * w         (wave per edge)
  k_aggr<<<(N_EDGES + WPB - 1) / WPB, B, 0, stream>>>(ei, ew, xh, aggr);

  // 7) xw = xh @ arma_W   (WMMA; wave owns 4 of 8 col tiles; W via TDM)
  {
    const int total = (N_NODES / 16) * 2;              // TGROUPS = 128/16/4 = 2
    size_t shmem = (size_t)HID * OUT * sizeof(float);  // 48 KB
    k_gemm<HID, OUT, 4, false>
        <<<(total + WPB - 1) / WPB, B, shmem, stream>>>(xh, aW, nullptr, xw);
  }

  // 8) prop[dst] += xw[src] * dinv[s]*w*dinv[d]
  k_prop<<<(N_EDGES + WPB - 1) / WPB, B, 0, stream>>>(ei, ew, dinv, xw, prop);

  // 9) fused sage/arma GEMMs + activations -> o3 (3 weights via TDM, 144 KB LDS)
  {
    const int total = (N_NODES / 16) * 2;
    size_t shmem = 3 * (size_t)HID * OUT * sizeof(float);
    k_final<<<(total + WPB - 1) / WPB, B, shmem, stream>>>(
        xh, aggr, cnt, prop, sWl, sbl, sWr, aV, abv, o3);
  }
}